// IW_MaxSquareloss_19292993093662
// MI455X (gfx1250) — compile-verified
//
#include <hip/hip_runtime.h>

// IW_MaxSquareloss for pred[N=4, C=19, H=512, W=1024] f32.
// loss = -(1/(N*C)) * sum_{n,c} w[n,c] * S[n,c]
//   S[n,c] = sum over pixels with argmax==c of sum_k softmax(x)_k^2
//   w[n,c] = 1 / max(cnt[n,c]^0.2 * (H*W)^0.8, 1)
// HBM-bound single pass (~159MB -> ~6.8us floor @ 23.3 TB/s). The 19-channel
// gather (row stride 2MB) is staged to LDS by the Tensor Data Mover on the
// CDNA5 path; a direct-load kernel remains as the guarded fallback.

#define C_CH   19
#define HW_LOG 19              // H*W = 512*1024 = 2^19
#define HW_SZ  (1 << HW_LOG)   // 524288
#define N_S    4
#define RATIO_F 0.2f
#define TILE   512             // pixels per TDM tile
#define BLK    512             // threads per block (16 waves)

#if __has_builtin(__builtin_amdgcn_tensor_load_to_lds) && \
    __has_builtin(__builtin_amdgcn_s_wait_tensorcnt)
#define USE_TDM 1
#else
#define USE_TDM 0
#endif

// ---------------------------------------------------------------- init / math

__global__ __launch_bounds__(256) void msq_init_ws(float* __restrict__ S,
                                                   unsigned* __restrict__ CNT) {
    int i = threadIdx.x;
    if (i < N_S * C_CH) { S[i] = 0.0f; CNT[i] = 0u; }
}

// Per-pixel softmax^2 sum + argmax, accumulated into LDS class bins.
__device__ __forceinline__ void msq_pixel(const float* __restrict__ col, int stride,
                                          float* __restrict__ ls,
                                          unsigned* __restrict__ lc) {
    float x[C_CH];
#pragma unroll
    for (int c = 0; c < C_CH; ++c) x[c] = col[(size_t)c * stride];

    float m = x[0];
    int   a = 0;
#pragma unroll
    for (int c = 1; c < C_CH; ++c) {
        if (x[c] > m) { m = x[c]; a = c; }   // first-max tie-break == jnp.argmax
    }
    float denom = 0.0f, sq = 0.0f;
#pragma unroll
    for (int c = 0; c < C_CH; ++c) {
        float e = __expf(x[c] - m);
        denom += e;
        sq    += e * e;
    }
    atomicAdd(&ls[a], sq / (denom * denom));
    atomicAdd(&lc[a], 1u);
}

// ---------------------------------------------------------------- TDM kernel

#if USE_TDM
typedef unsigned int v4u __attribute__((ext_vector_type(4)));
typedef int          v8i __attribute__((ext_vector_type(8)));
typedef int          v4i __attribute__((ext_vector_type(4)));

// Issue one TDM 2D tile load: 19 rows x TILE floats, row stride HW floats,
// packed contiguously to LDS at lds_byte_off. ISA cdna5 sec 8.3/8.4 layout.
// This toolchain's builtin takes 6 args (g0 x4, g1 x8, g2 x4, g3 x4, x8, cpol);
// trailing groups are zero = "unused" for a 2D tile.
__device__ __forceinline__ void tdm_load_tile(const void* gptr, unsigned lds_byte_off) {
    unsigned long long ga = (unsigned long long)(uintptr_t)gptr;
    v4u g0;
    g0.x = 1u;                                           // count=1, no gather
    g0.y = lds_byte_off;                                 // lds_addr (bytes)
    g0.z = (unsigned)ga;                                 // global_addr[31:0]
    g0.w = ((unsigned)(ga >> 32) & 0x01FFFFFFu)          // global_addr[56:32]
         | 0x80000000u;                                  // type=2 ("image")
    v8i g1;
    g1[0] = 0x20000;                                     // data_size=2 (4B)
    g1[1] = 0;                                           // tensor_dim0[15:0]=0 (HW lo16)
    g1[2] = (int)((C_CH << 16) | (HW_SZ >> 16));         // dim0 hi16 | tensor_dim1=19
    g1[3] = (int)(TILE << 16);                           // dim1 hi16=0 | tile_dim0=TILE
    g1[4] = C_CH;                                        // tile_dim1=19 | tile_dim2=0
    g1[5] = HW_SZ;                                       // tensor_dim0_stride lo32
    g1[6] = 0;                                           // stride hi16 | dim1_stride lo16
    g1[7] = 0;
    v4i z4 = {0, 0, 0, 0};                               // groups 2/3 unused (2D tile)
    v8i z8 = {0, 0, 0, 0, 0, 0, 0, 0};                   // trailing group unused
    __builtin_amdgcn_tensor_load_to_lds(g0, g1, z4, z4, z8, 0);
}

__global__ __launch_bounds__(BLK) void msq_main_tdm(const float* __restrict__ pred,
                                                    float* __restrict__ S,
                                                    unsigned* __restrict__ CNT) {
    __shared__ float    buf[2][C_CH][TILE];              // double-buffered tiles
    __shared__ float    ls[C_CH];
    __shared__ unsigned lc[C_CH];

    const int tid = threadIdx.x;
    if (tid < C_CH) { ls[tid] = 0.0f; lc[tid] = 0u; }

    // Block covers 2*TILE consecutive pixels; never straddles a sample n.
    const long long base = (long long)blockIdx.x * (2 * TILE);
    const int n  = (int)(base >> HW_LOG);
    const int p0 = (int)(base & (HW_SZ - 1));
    const float* gbase = pred + (size_t)n * C_CH * HW_SZ + (size_t)p0;

    // TDM ignores EXEC -> must be a *scalar* branch so only wave 0 issues.
    if (__builtin_amdgcn_readfirstlane((int)(threadIdx.x >> 5)) == 0) {
        tdm_load_tile(gbase,        (unsigned)(uintptr_t)&buf[0][0][0]);
        tdm_load_tile(gbase + TILE, (unsigned)(uintptr_t)&buf[1][0][0]);
    }
    // Issuing wave waits for tile0 (in-order TDM); other waves pass (cnt==0).
    __builtin_amdgcn_s_wait_tensorcnt(1);
    __syncthreads();

    msq_pixel(&buf[0][0][tid], TILE, ls, lc);            // compute tile 0

    __builtin_amdgcn_s_wait_tensorcnt(0);                // tile1 complete
    __syncthreads();

    msq_pixel(&buf[1][0][tid], TILE, ls, lc);            // compute tile 1
    __syncthreads();

    if (tid < C_CH) {
        atomicAdd(&S[n * C_CH + tid], ls[tid]);
        atomicAdd(&CNT[n * C_CH + tid], lc[tid]);
    }
}
#endif // USE_TDM

// ------------------------------------------------------- fallback direct-load

__global__ __launch_bounds__(256) void msq_main_direct(const float* __restrict__ pred,
                                                       float* __restrict__ S,
                                                       unsigned* __restrict__ CNT) {
    __shared__ float    ls[C_CH];
    __shared__ unsigned lc[C_CH];

    const int tid = threadIdx.x;
    if (tid < C_CH) { ls[tid] = 0.0f; lc[tid] = 0u; }
    __syncthreads();

    const long long gid = (long long)blockIdx.x * 256 + tid;
    const int n = (int)(gid >> HW_LOG);
    const int p = (int)(gid & (HW_SZ - 1));
    const float* base = pred + (size_t)n * C_CH * HW_SZ + (size_t)p;

    __builtin_prefetch(base + 16384, 0, 0);              // global_prefetch_b8
    msq_pixel(base, HW_SZ, ls, lc);
    __syncthreads();

    if (tid < C_CH) {
        atomicAdd(&S[n * C_CH + tid], ls[tid]);
        atomicAdd(&CNT[n * C_CH + tid], lc[tid]);
    }
}

// ------------------------------------------------------------------- finalize

__global__ __launch_bounds__(128) void msq_final(const float* __restrict__ S,
                                                 const unsigned* __restrict__ CNT,
                                                 float* __restrict__ out) {
    __shared__ float red[128];
    const int i = threadIdx.x;
    float v = 0.0f;
    if (i < N_S * C_CH) {
        float cnt = (float)CNT[i];
        float w = 1.0f / fmaxf(powf(cnt, RATIO_F) * powf((float)HW_SZ, 1.0f - RATIO_F), 1.0f);
        v = S[i] * w;
    }
    red[i] = v;
    __syncthreads();
#pragma unroll
    for (int s = 64; s > 0; s >>= 1) {
        if (i < s) red[i] += red[i + s];
        __syncthreads();
    }
    if (i == 0) out[0] = -red[0] / (float)(N_S * C_CH);
}

// --------------------------------------------------------------------- launch

extern "C" void kernel_launch(void* const* d_in, const int* in_sizes, int n_in,
                              void* d_out, int out_size, void* d_ws, size_t ws_size,
                              hipStream_t stream) {
    (void)in_sizes; (void)n_in; (void)out_size; (void)ws_size;
    const float* pred = (const float*)d_in[0];
    float*    S   = (float*)d_ws;                 // 76 floats
    unsigned* CNT = (unsigned*)(S + N_S * C_CH);  // 76 uints
    float*    out = (float*)d_out;

    msq_init_ws<<<1, 256, 0, stream>>>(S, CNT);
#if USE_TDM
    msq_main_tdm<<<(N_S * HW_SZ) / (2 * TILE), BLK, 0, stream>>>(pred, S, CNT);
#else
    msq_main_direct<<<(N_S * HW_SZ) / 256, 256, 0, stream>>>(pred, S, CNT);
#endif
    msq_final<<<1, 128, 0, stream>>>(S, CNT, out);
}